// GraphConvolution_64192581206537
// MI455X (gfx1250) — compile-verified
//
#include <hip/hip_runtime.h>

// ---------------------------------------------------------------------------
// GraphConvolution (Gaussian-mixture spectral conv) for MI455X / gfx1250.
//
// Phase 1 (WMMA GEMM): support[N][256] = x[N][64] @ weight[64][256]
//   weight [IN_F][OUT_F][KER] flat == [64][256] row-major, and C's column
//   index c = o*4+k gives exactly the [n][o][k] layout the edge phase wants.
//   1.64 GFLOP -> negligible; done with V_WMMA_F32_16X16X4_F32.
//   Row bounds handled by a once-per-tile clamp (not a per-load guard) so the
//   K-loop is branchless: 16 straight-line b64/b32 loads + 16 v_wmma.
//
// Phase 2 (edge scatter): out[row] += sum_k value[e,k] * support[col][:,k]
//   Folding the K-sum into the message makes this [N,64] accumulation
//   (4x less atomic traffic than the reference's [N,64,4] segment_sum).
//   support (51.2 MB) + out (12.8 MB) fit in the 192 MB L2, so the ~1 GB of
//   random gather + f32-atomic traffic stays on-chip at L2 speed.
// ---------------------------------------------------------------------------

typedef __attribute__((ext_vector_type(2))) float v2f;
typedef __attribute__((ext_vector_type(8))) float v8f;

#define IN_F  64
#define OUT_F 64
#define KER   4
#define EMB   3
#define CDIM  (OUT_F * KER)   // 256

// ---- Phase 1: support = x @ weight via V_WMMA_F32_16X16X4_F32 --------------
// One wave computes one 16x16 f32 tile of C; K loop steps by 4.
__global__ void support_gemm_wmma(const float* __restrict__ x,
                                  const float* __restrict__ w,
                                  float* __restrict__ support,
                                  int N) {
  const int lane = threadIdx.x & 31;
  const int wave = (blockIdx.x * blockDim.x + threadIdx.x) >> 5;
  const int waveStride = (gridDim.x * blockDim.x) >> 5;

  const int tilesPerRow = CDIM / 16;              // 16 column tiles
  const int rowTiles    = (N + 15) >> 4;
  const int totalTiles  = rowTiles * tilesPerRow;

  const int half = lane >> 4;    // 0: lanes 0-15, 1: lanes 16-31
  const int lid  = lane & 15;

  for (int t = wave; t < totalTiles; t += waveStride) {
    const int m0 = (t / tilesPerRow) * 16;
    const int c0 = (t % tilesPerRow) * 16;

    // Clamp the A row ONCE per tile: all K-loop loads become unconditional
    // (clamped rows load valid memory; their stores are guarded below).
    int arow = m0 + lid;
    if (arow >= N) arow = N - 1;
    const float* __restrict__ arowp = x + (size_t)arow * IN_F + half * 2;
    const float* __restrict__ bcolp = w + c0 + lid + (size_t)(half * 2) * CDIM;

    v8f acc = {};
#pragma unroll
    for (int k0 = 0; k0 < IN_F; k0 += 4) {
      // A fragment (16x4 f32): lanes 0-15 hold K={k0,k0+1}, lanes 16-31 K={k0+2,k0+3}
      v2f a;
      a.x = arowp[k0];
      a.y = arowp[k0 + 1];
      // B fragment (4x16 f32, KxN), mirrored layout: N=lid per lane,
      // lanes 0-15 hold rows K={k0,k0+1}, lanes 16-31 rows K={k0+2,k0+3}
      v2f b;
      b.x = bcolp[(size_t)k0 * CDIM];
      b.y = bcolp[(size_t)(k0 + 1) * CDIM];
      acc = __builtin_amdgcn_wmma_f32_16x16x4_f32(
          /*neg_a=*/false, a, /*neg_b=*/false, b,
          /*c_mod=*/(short)0, acc, /*reuse_a=*/false, /*reuse_b=*/false);
    }
    // C/D layout: VGPR r -> M = m0 + r + 8*half, N = c0 + lid
#pragma unroll
    for (int r = 0; r < 8; ++r) {
      const int row = m0 + r + 8 * half;
      if (row < N) support[(size_t)row * CDIM + c0 + lid] = acc[r];
    }
  }
}

// ---- Phase 1b: out = bias (broadcast) --------------------------------------
__global__ void init_out(float* __restrict__ out,
                         const float* __restrict__ bias, int total) {
  const int i = blockIdx.x * blockDim.x + threadIdx.x;
  if (i < total) out[i] = bias[i & (OUT_F - 1)];
}

// ---- Phase 2: per-edge Gaussian weights + gather/scatter -------------------
// One wave per edge (grid-stride). Each lane handles outputs {2*lane, 2*lane+1}:
// two coalesced b128 loads cover support[col][o][0..3], then two hardware
// global_atomic_add_f32 into out[row].
__global__ void edge_scatter(const int*   __restrict__ eidx,     // [2*E]
                             const float* __restrict__ spec,     // [N][3]
                             const float* __restrict__ mu,       // [3][4]
                             const float* __restrict__ sig,      // [4]
                             const float* __restrict__ support,  // [N][64][4]
                             float* __restrict__ out,            // [N][64]
                             int E) {
  const int lane = threadIdx.x & 31;
  const int wave = (blockIdx.x * blockDim.x + threadIdx.x) >> 5;
  const int waveStride = (gridDim.x * blockDim.x) >> 5;

  // Hoist the tiny constant tables (12 + 4 floats).
  float mu0[KER], mu1[KER], mu2[KER], sg[KER];
#pragma unroll
  for (int k = 0; k < KER; ++k) {
    mu0[k] = mu[0 * KER + k];
    mu1[k] = mu[1 * KER + k];
    mu2[k] = mu[2 * KER + k];
    sg[k]  = sig[k];
  }

  for (int e = wave; e < E; e += waveStride) {
    const int row = eidx[e];
    const int col = eidx[E + e];

    // diff = spec[row] - spec[col]  (uniform across the wave -> broadcast loads)
    const float d0 = spec[row * EMB + 0] - spec[col * EMB + 0];
    const float d1 = spec[row * EMB + 1] - spec[col * EMB + 1];
    const float d2 = spec[row * EMB + 2] - spec[col * EMB + 2];

    float val[KER];
#pragma unroll
    for (int k = 0; k < KER; ++k) {
      const float a0 = d0 - mu0[k];
      const float a1 = d1 - mu1[k];
      const float a2 = d2 - mu2[k];
      const float qq = -0.5f * (a0 * a0 + a1 * a1 + a2 * a2);
      val[k] = __expf(sg[k] * qq);
    }

    // Gather support[col]: lane covers outputs 2*lane and 2*lane+1, k-minor.
    const float4* sp = (const float4*)(support + (size_t)col * CDIM + lane * 8);
    const float4 s0 = sp[0];
    const float4 s1 = sp[1];
    const float m0 = s0.x * val[0] + s0.y * val[1] + s0.z * val[2] + s0.w * val[3];
    const float m1 = s1.x * val[0] + s1.y * val[1] + s1.z * val[2] + s1.w * val[3];

    float* dst = out + (size_t)row * OUT_F + lane * 2;
    unsafeAtomicAdd(dst,     m0);   // global_atomic_add_f32
    unsafeAtomicAdd(dst + 1, m1);
  }
}

// ---------------------------------------------------------------------------
extern "C" void kernel_launch(void* const* d_in, const int* in_sizes, int n_in,
                              void* d_out, int out_size, void* d_ws, size_t ws_size,
                              hipStream_t stream) {
  const float* x    = (const float*)d_in[0];
  const int*   eidx = (const int*)  d_in[1];
  const float* spec = (const float*)d_in[2];
  const float* w    = (const float*)d_in[3];
  const float* bias = (const float*)d_in[4];
  const float* mu   = (const float*)d_in[5];
  const float* sig  = (const float*)d_in[6];

  const int N = in_sizes[0] / IN_F;
  const int E = in_sizes[1] / 2;

  float* support = (float*)d_ws;   // N * 256 floats (51.2 MB for N=50000)
  float* out     = (float*)d_out;  // N * 64 floats

  // Phase 1: WMMA GEMM (stream-ordered before the edge phase reads support)
  {
    const int totalTiles = ((N + 15) / 16) * (CDIM / 16);
    const int wavesPerBlock = 8;   // 256 threads
    const int blocks = (totalTiles + wavesPerBlock - 1) / wavesPerBlock;
    support_gemm_wmma<<<blocks, 256, 0, stream>>>(x, w, support, N);
  }

  // Phase 1b: seed out with the bias
  {
    const int total = N * OUT_F;
    init_out<<<(total + 255) / 256, 256, 0, stream>>>(out, bias, total);
  }

  // Phase 2: edge gather/scale/scatter (grid-stride over E edges, 1 wave/edge)
  {
    const int blocks = 2048;       // 16384 waves over 800k edges
    edge_scatter<<<blocks, 256, 0, stream>>>(eidx, spec, mu, sig, support, out, E);
  }
}